// GrGyroBNImpl_50766513438861
// MI455X (gfx1250) — compile-verified
//
#include <hip/hip_runtime.h>
#include <hip/hip_bf16.h>
#include <math.h>
#include <stdint.h>

#define DEV static __device__ __forceinline__

#ifndef __has_builtin
#define __has_builtin(x) 0
#endif

typedef __attribute__((ext_vector_type(2))) float v2f;
typedef __attribute__((ext_vector_type(8))) float v8f;

static constexpr int Bn = 256, Cn = 128, Nn = 64, Pn = 16;
static constexpr float EPSf = 1e-5f;
static constexpr int JACOBI_SWEEPS = 6;

// ---------------- CDNA5 async global<->LDS (ASYNCcnt path) -----------------
// Builtin signature (from clang): (int4 AS1* gaddr, int4 AS3* lds, Ii, Ii)
typedef int v4i_vs __attribute__((vector_size(16)));
typedef __attribute__((address_space(1))) v4i_vs GV4;  // global int4
typedef __attribute__((address_space(3))) v4i_vs LV4;  // LDS int4 (32-bit ptr)

DEV void g2l_async_b128(float* ldsDst, const float* gSrc) {
#if __has_builtin(__builtin_amdgcn_global_load_async_to_lds_b128)
  __builtin_amdgcn_global_load_async_to_lds_b128(
      (GV4*)(uintptr_t)gSrc, (LV4*)(uint32_t)(uintptr_t)ldsDst, 0, 0);
#else
  const uint32_t lo = (uint32_t)(uintptr_t)ldsDst;  // low 32 bits = LDS byte addr
  asm volatile("global_load_async_to_lds_b128 %0, %1, off"
               :: "v"(lo), "v"(gSrc) : "memory");
#endif
}

DEV void l2g_async_b128(float* gDst, const float* ldsSrc) {
#if __has_builtin(__builtin_amdgcn_global_store_async_from_lds_b128)
  __builtin_amdgcn_global_store_async_from_lds_b128(
      (GV4*)(uintptr_t)gDst, (LV4*)(uint32_t)(uintptr_t)ldsSrc, 0, 0);
#else
  const uint32_t lo = (uint32_t)(uintptr_t)ldsSrc;
  asm volatile("global_store_async_from_lds_b128 %0, %1, off"
               :: "v"(gDst), "v"(lo) : "memory");
#endif
}

DEV void wait_async0() {
#if __has_builtin(__builtin_amdgcn_s_wait_asynccnt)
  __builtin_amdgcn_s_wait_asynccnt(0);
#else
  asm volatile("s_wait_asynccnt 0x0" ::: "memory");
#endif
}

// n floats, n % 128 == 0; 32 lanes x b128 per step.
DEV void stage_in(float* ldsDst, const float* gSrc, int n, int lane) {
  for (int i = lane * 4; i < n; i += 128) g2l_async_b128(ldsDst + i, gSrc + i);
}
DEV void stage_out(float* gDst, const float* ldsSrc, int n, int lane) {
  for (int i = lane * 4; i < n; i += 128) l2g_async_b128(gDst + i, ldsSrc + i);
}

// ---------------- WMMA core: D = A(16x4) * B(4x16) + C, f32 ----------------
DEV v8f wmma4(v2f a, v2f b, v8f c) {
  return __builtin_amdgcn_wmma_f32_16x16x4_f32(
      /*neg_a=*/false, a, /*neg_b=*/false, b,
      /*c_mod=*/(short)0, c, /*reuse_a=*/false, /*reuse_b=*/false);
}

DEV v8f zero8() {
  v8f z;
#pragma unroll
  for (int r = 0; r < 8; ++r) z[r] = 0.f;
  return z;
}

// One 16x16 output tile: acc += A(16xK) * B(Kx16), K multiple of 4.
// transA: A stored KxM (reads A[k*lda+m]); transB: B stored NxK (reads B[n*ldb+k]).
DEV v8f gemm_tile16(const float* A, int lda, bool transA,
                    const float* Bm, int ldb, bool transB,
                    int K, v8f acc, int lane) {
  const int mn = lane & 15;           // M index for A frag, N index for B frag
  const int kh = (lane >> 4) << 1;    // lanes 16-31 hold K+2,K+3
  for (int k0 = 0; k0 < K; k0 += 4) {
    const int k1 = k0 + kh, k2 = k0 + kh + 1;
    v2f a, b;
    a.x = transA ? A[k1 * lda + mn] : A[mn * lda + k1];
    a.y = transA ? A[k2 * lda + mn] : A[mn * lda + k2];
    b.x = transB ? Bm[mn * ldb + k1] : Bm[k1 * ldb + mn];
    b.y = transB ? Bm[mn * ldb + k2] : Bm[k2 * ldb + mn];
    acc = wmma4(a, b, acc);
  }
  return acc;
}

// C/D 16x16 layout: VGPR r -> row r (lanes 0-15) / row r+8 (lanes 16-31)
DEV void store_c(float* C, int ldc, v8f acc, int lane) {
  const int n = lane & 15;
  const int half = lane >> 4;
#pragma unroll
  for (int r = 0; r < 8; ++r) C[(r + 8 * half) * ldc + n] = acc[r];
}

// C(Mx16) = A(MxK) * B(Kx16); M multiple of 16.
DEV void wave_gemm(const float* A, int lda, bool transA,
                   const float* Bm, int ldb, bool transB,
                   float* C, int ldc, int M, int K, int lane) {
  for (int m0 = 0; m0 < M; m0 += 16) {
    const float* Ablk = transA ? (A + m0) : (A + (size_t)m0 * lda);
    v8f acc = zero8();
    acc = gemm_tile16(Ablk, lda, transA, Bm, ldb, transB, K, acc, lane);
    store_c(C + (size_t)m0 * ldc, ldc, acc, lane);
  }
}

// ---------------- cooperative 16x16 Jacobi eigensolver (symmetric) ----------
DEV void jacobi16(float* G, float* V, float* S, int lane) {
  if (lane < 16)
    for (int j = 0; j < 16; ++j) V[lane * 16 + j] = (lane == j) ? 1.f : 0.f;
  __syncthreads();
  for (int sweep = 0; sweep < JACOBI_SWEEPS; ++sweep) {
    for (int p = 0; p < 15; ++p) {
      for (int q = p + 1; q < 16; ++q) {
        const float apq = G[p * 16 + q];
        const float app = G[p * 16 + p];
        const float aqq = G[q * 16 + q];
        float c = 1.f, s = 0.f;
        if (fabsf(apq) > 1e-12f) {
          const float tau = (aqq - app) / (2.f * apq);
          const float t = copysignf(1.f, tau) / (fabsf(tau) + sqrtf(1.f + tau * tau));
          c = rsqrtf(1.f + t * t);
          s = t * c;
        }
        __syncthreads();
        if (lane < 16) {  // column rotation
          const int i = lane;
          const float gip = G[i * 16 + p], giq = G[i * 16 + q];
          G[i * 16 + p] = c * gip - s * giq;
          G[i * 16 + q] = s * gip + c * giq;
        }
        __syncthreads();
        if (lane < 16) {  // row rotation + accumulate V
          const int i = lane;
          const float gpi = G[p * 16 + i], gqi = G[q * 16 + i];
          G[p * 16 + i] = c * gpi - s * gqi;
          G[q * 16 + i] = s * gpi + c * gqi;
          const float vip = V[i * 16 + p], viq = V[i * 16 + q];
          V[i * 16 + p] = c * vip - s * viq;
          V[i * 16 + q] = s * vip + c * viq;
        }
        __syncthreads();
      }
    }
  }
  if (lane < 16) S[lane] = sqrtf(fmaxf(G[lane * 16 + lane], 0.f));
  __syncthreads();
}

// ---------------- cooperative 16x16 Gauss-Jordan inverse --------------------
DEV void inv16(const float* Min, float* aug, float* Mout, int lane) {
  if (lane < 16)
    for (int j = 0; j < 16; ++j) {
      aug[lane * 32 + j] = Min[lane * 16 + j];
      aug[lane * 32 + 16 + j] = (lane == j) ? 1.f : 0.f;
    }
  __syncthreads();
  for (int k = 0; k < 16; ++k) {
    float piv = aug[k * 32 + k];
    piv = (fabsf(piv) < 1e-20f) ? 1e-20f : piv;
    const float ip = 1.f / piv;
    __syncthreads();
    aug[k * 32 + lane] *= ip;
    __syncthreads();
    if (lane < 16 && lane != k) {
      const float f = aug[lane * 32 + k];
      for (int j = 0; j < 32; ++j) aug[lane * 32 + j] -= f * aug[k * 32 + j];
    }
    __syncthreads();
  }
  if (lane < 16)
    for (int j = 0; j < 16; ++j) Mout[lane * 16 + j] = aug[lane * 32 + 16 + j];
  __syncthreads();
}

// ---------------- thin SVD of A (Mx16): A = Q diag(S) V^T -------------------
DEV void svd_tall(const float* A, int M, float* G, float* V, float* S,
                  float* Q, int lane) {
  wave_gemm(A, 16, true, A, 16, false, G, 16, 16, M, lane);  // G = A^T A
  __syncthreads();
  jacobi16(G, V, S, lane);
  wave_gemm(A, 16, false, V, 16, false, Q, 16, M, 16, lane); // Q = A V
  __syncthreads();
  for (int i = lane; i < M * 16; i += 32)
    Q[i] = Q[i] / fmaxf(S[i & 15], 1e-12f);
  __syncthreads();
}

// ---------------- cooperative 64x64 Gauss-Jordan (augmented 64x128) ---------
DEV void gj64(float* sL, int lane) {
  for (int k = 0; k < 64; ++k) {
    float piv = sL[k * 128 + k];
    piv = (fabsf(piv) < 1e-20f) ? 1e-20f : piv;
    const float ip = 1.f / piv;
    __syncthreads();
    for (int j = lane; j < 128; j += 32) sL[k * 128 + j] *= ip;
    __syncthreads();
#pragma unroll
    for (int rr = 0; rr < 2; ++rr) {
      const int r = lane + rr * 32;
      if (r != k) {
        const float f = sL[r * 128 + k];
        for (int j = 0; j < 128; ++j) sL[r * 128 + j] -= f * sL[k * 128 + j];
      }
    }
    __syncthreads();
  }
}

// ============================================================================
// K1: per-(b,c) log_map(mean0, X) -> xlog staged into out X-region
// ============================================================================
__global__ __launch_bounds__(32)
void k_karcher_log(const float* __restrict__ X, float* __restrict__ xlogOut) {
  const int bc = blockIdx.x;
  const int c = bc % Cn;
  const int lane = threadIdx.x;
  __shared__ alignas(16) float sX[1024], sU[1024], sT[1024];
  __shared__ float sUtX[256], sInv[256], sAug[512],
                   sG[256], sV[256], sS[16], sQ[1024], sA[1024];
  stage_in(sX, X + (size_t)bc * 1024, 1024, lane);
  stage_in(sU, X + (size_t)c * 1024, 1024, lane);   // mean = X[0, c]
  wait_async0();
  __syncthreads();
  wave_gemm(sU, 16, true, sX, 16, false, sUtX, 16, 16, 64, lane);  // U^T X
  __syncthreads();
  inv16(sUtX, sAug, sInv, lane);
  wave_gemm(sU, 16, false, sUtX, 16, false, sT, 16, 64, 16, lane); // U (U^T X)
  __syncthreads();
  for (int i = lane; i < 1024; i += 32) sA[i] = sX[i] - sT[i];
  __syncthreads();
  wave_gemm(sA, 16, false, sInv, 16, false, sT, 16, 64, 16, lane); // (X-UUtX) inv
  __syncthreads();
  svd_tall(sT, 64, sG, sV, sS, sQ, lane);
  for (int i = lane; i < 1024; i += 32) sA[i] = sQ[i] * atanf(sS[i & 15]);
  __syncthreads();
  wave_gemm(sA, 16, false, sV, 16, true, sT, 16, 64, 16, lane);    // Q atan(S) V^T
  __syncthreads();
  stage_out(xlogOut + (size_t)bc * 1024, sT, 1024, lane);  // completes by ENDPGM
}

// ============================================================================
// K2: per-channel batch mean of xlog + exp_map -> updated mean
// ============================================================================
__global__ __launch_bounds__(32)
void k_expmap(const float* __restrict__ X, const float* __restrict__ xlogIn,
              float* __restrict__ meanOut) {
  const int c = blockIdx.x;
  const int lane = threadIdx.x;
  __shared__ alignas(16) float sU[1024];
  __shared__ float sD[1024], sG[256], sV[256], sS[16],
                   sQ[1024], sT[1024], sW[256], sT16[256], sM2[1024];
  stage_in(sU, X + (size_t)c * 1024, 1024, lane);
  const float invB = 1.f / (float)Bn;
  for (int i = lane; i < 1024; i += 32) {
    float acc = 0.f;
    for (int b = 0; b < Bn; ++b)
      acc += xlogIn[((size_t)b * Cn + c) * 1024 + i];
    sD[i] = acc * invB;
  }
  wait_async0();
  __syncthreads();
  svd_tall(sD, 64, sG, sV, sS, sQ, lane);          // D = Q diag(S) V^T
  for (int i = lane; i < 256; i += 32) sW[i] = sV[i] * cosf(sS[i & 15]);
  __syncthreads();
  wave_gemm(sW, 16, false, sV, 16, true, sT16, 16, 16, 16, lane);  // V cos V^T
  __syncthreads();
  wave_gemm(sU, 16, false, sT16, 16, false, sT, 16, 64, 16, lane); // U (V cos V^T)
  __syncthreads();
  for (int i = lane; i < 1024; i += 32) sD[i] = sQ[i] * sinf(sS[i & 15]);
  __syncthreads();
  wave_gemm(sD, 16, false, sV, 16, true, sM2, 16, 64, 16, lane);   // Q sin V^T
  __syncthreads();
  for (int i = lane; i < 1024; i += 32)
    meanOut[(size_t)c * 1024 + i] = sT[i] + sM2[i];
}

// ============================================================================
// K3: per-(b,c) principal-angle energy: sum(theta^2) of svd(mean^T X)
// ============================================================================
__global__ __launch_bounds__(32)
void k_var1(const float* __restrict__ X, const float* __restrict__ meanIn,
            float* __restrict__ varPart) {
  const int bc = blockIdx.x;
  const int c = bc % Cn;
  const int lane = threadIdx.x;
  __shared__ alignas(16) float sX[1024], sU[1024];
  __shared__ float sM[256], sG[256], sV[256], sS[16], red[32];
  stage_in(sX, X + (size_t)bc * 1024, 1024, lane);
  stage_in(sU, meanIn + (size_t)c * 1024, 1024, lane);
  wait_async0();
  __syncthreads();
  wave_gemm(sU, 16, true, sX, 16, false, sM, 16, 16, 64, lane);  // mean^T X
  __syncthreads();
  wave_gemm(sM, 16, true, sM, 16, false, sG, 16, 16, 16, lane);  // M^T M
  __syncthreads();
  jacobi16(sG, sV, sS, lane);
  float th2 = 0.f;
  if (lane < 16) {
    const float s = fminf(fmaxf(sS[lane], -1.f), 1.f);
    const float th = acosf(s);
    th2 = th * th;
  }
  red[lane] = th2;
  __syncthreads();
  if (lane == 0) {
    float acc = 0.f;
    for (int j = 0; j < 16; ++j) acc += red[j];
    varPart[bc] = acc;
  }
}

// ============================================================================
// K4: per-channel: var reduce, Bm = log_identity_B(mean), Rinv = cayley(-skew)
// ============================================================================
__global__ __launch_bounds__(32)
void k_channel(const float* __restrict__ meanIn, const float* __restrict__ varPart,
               float* __restrict__ var, float* __restrict__ Rinv) {
  const int c = blockIdx.x;
  const int lane = threadIdx.x;
  __shared__ alignas(16) float sM[1024];
  __shared__ alignas(16) float sL[64 * 128];
  __shared__ float sAug[512], sIu[256], sCm[768], sG[256], sV[256],
                   sS[16], sQ[768], sSc[768], sBm[768], red[32];
  stage_in(sM, meanIn + (size_t)c * 1024, 1024, lane);
  // variance: deterministic fixed-order reduction
  float acc = 0.f;
  for (int b = lane; b < Bn; b += 32) acc += varPart[(size_t)b * Cn + c];
  red[lane] = acc;
  wait_async0();
  __syncthreads();
  if (lane == 0) {
    float t = 0.f;
    for (int j = 0; j < 32; ++j) t += red[j];
    var[c] = t / (float)Bn;
  }
  inv16(sM, sAug, sIu, lane);                                     // inv(mean[:16])
  wave_gemm(sM + 256, 16, false, sIu, 16, false, sCm, 16, 48, 16, lane);
  __syncthreads();
  svd_tall(sCm, 48, sG, sV, sS, sQ, lane);
  for (int i = lane; i < 768; i += 32) sSc[i] = sQ[i] * atanf(sS[i & 15]);
  __syncthreads();
  wave_gemm(sSc, 16, false, sV, 16, true, sBm, 16, 48, 16, lane); // Bm (48x16)
  __syncthreads();
  // Augmented [I + S/2 | I - S/2] for cayley(-b2skew(Bm))
  for (int i = lane; i < 64 * 128; i += 32) {
    const int r = i >> 7, cc = i & 127;
    const int j = cc & 63;
    const bool rhs = cc >= 64;
    float s = 0.f;
    if (r < 16 && j >= 16)      s = -sBm[(j - 16) * 16 + r];
    else if (r >= 16 && j < 16) s =  sBm[(r - 16) * 16 + j];
    sL[r * 128 + cc] = ((r == j) ? 1.f : 0.f) + (rhs ? -0.5f : 0.5f) * s;
  }
  __syncthreads();
  gj64(sL, lane);
  for (int i = lane * 4; i < 4096; i += 128)   // rows contiguous in 16B chunks
    l2g_async_b128(Rinv + (size_t)c * 4096 + i, &sL[(i >> 6) * 128 + 64 + (i & 63)]);
}

// ============================================================================
// K5: Rb = cayley(b2skew(bias)) (single block)
// ============================================================================
__global__ __launch_bounds__(32)
void k_rb(const float* __restrict__ bias, float* __restrict__ Rb) {
  const int lane = threadIdx.x;
  __shared__ alignas(16) float sL[64 * 128];
  __shared__ float sB[768];
  for (int i = lane; i < 768; i += 32) sB[i] = bias[i];
  __syncthreads();
  for (int i = lane; i < 64 * 128; i += 32) {
    const int r = i >> 7, cc = i & 127;
    const int j = cc & 63;
    const bool rhs = cc >= 64;
    float s = 0.f;
    if (r < 16 && j >= 16)      s = -sB[(j - 16) * 16 + r];
    else if (r >= 16 && j < 16) s =  sB[(r - 16) * 16 + j];
    sL[r * 128 + cc] = ((r == j) ? 1.f : 0.f) + (rhs ? 0.5f : -0.5f) * s;
  }
  __syncthreads();
  gj64(sL, lane);
  for (int i = lane * 4; i < 4096; i += 128)
    l2g_async_b128(Rb + i, &sL[(i >> 6) * 128 + 64 + (i & 63)]);
}

// ============================================================================
// K6: per-(b,c): center, gyro scalar-product, re-bias, write output
// ============================================================================
__global__ __launch_bounds__(32)
void k_final(const float* __restrict__ X, const float* __restrict__ Rinv,
             const float* __restrict__ Rb, const float* __restrict__ var,
             const float* __restrict__ shift, float* __restrict__ out) {
  const int bc = blockIdx.x;
  const int c = bc % Cn;
  const int lane = threadIdx.x;
  __shared__ alignas(16) float sR[4096], sX[1024], sXc[1024];
  __shared__ float sAug[512], sIu[256], sCm[768],
                   sG[256], sV[256], sS[16], sQ[768], sW[256];
  stage_in(sR, Rinv + (size_t)c * 4096, 4096, lane);
  stage_in(sX, X + (size_t)bc * 1024, 1024, lane);
  wait_async0();
  __syncthreads();
  wave_gemm(sR, 64, false, sX, 16, false, sXc, 16, 64, 64, lane);  // Xc = Rinv X
  __syncthreads();
  stage_in(sR, Rb, 4096, lane);            // overlap Rb fetch with gyro SVD
  const float factor = shift[0] * rsqrtf(var[c] + EPSf);
  inv16(sXc, sAug, sIu, lane);                                     // inv(Xc[:16])
  wave_gemm(sXc + 256, 16, false, sIu, 16, false, sCm, 16, 48, 16, lane);
  __syncthreads();
  svd_tall(sCm, 48, sG, sV, sS, sQ, lane);
  // top = V diag(cos th) V^T ; bot = Q diag(sin th) V^T ; th = factor*atan(S)
  for (int i = lane; i < 256; i += 32)
    sW[i] = sV[i] * cosf(factor * atanf(sS[i & 15]));
  __syncthreads();
  wave_gemm(sW, 16, false, sV, 16, true, sX, 16, 16, 16, lane);    // rows 0..15
  __syncthreads();
  for (int i = lane; i < 768; i += 32)
    sCm[i] = sQ[i] * sinf(factor * atanf(sS[i & 15]));
  __syncthreads();
  wave_gemm(sCm, 16, false, sV, 16, true, sX + 256, 16, 48, 16, lane);
  wait_async0();                            // Rb now resident in sR
  __syncthreads();
  wave_gemm(sR, 64, false, sX, 16, false, sXc, 16, 64, 64, lane);  // Xn = Rb Xs
  __syncthreads();
  stage_out(out + (size_t)bc * 1024, sXc, 1024, lane);  // completes by ENDPGM
}

// ============================================================================
extern "C" void kernel_launch(void* const* d_in, const int* in_sizes, int n_in,
                              void* d_out, int out_size, void* d_ws, size_t ws_size,
                              hipStream_t stream) {
  (void)in_sizes; (void)n_in; (void)out_size; (void)ws_size;
  const float* X     = (const float*)d_in[0];   // (B,C,64,16)
  const float* bias  = (const float*)d_in[1];   // (48,16)
  const float* shift = (const float*)d_in[2];   // scalar
  float* out = (float*)d_out;
  float* meanOut = out + (size_t)Bn * Cn * Nn * Pn;   // mean region of output

  float* ws      = (float*)d_ws;
  float* varPart = ws;                            // B*C
  float* var     = varPart + (size_t)Bn * Cn;     // C
  float* Rinv    = var + Cn;                      // C*4096 (16B-aligned offset)
  float* Rb      = Rinv + (size_t)Cn * 4096;      // 4096

  // Stage 1: xlog per sample, staged in the (later overwritten) output region.
  k_karcher_log<<<Bn * Cn, 32, 0, stream>>>(X, out);
  // Stage 2: Karcher batch mean + exp_map -> updated mean (also final output).
  k_expmap<<<Cn, 32, 0, stream>>>(X, out, meanOut);
  // Stage 3: principal-angle energies, per-channel variance + Cayley rotations.
  k_var1<<<Bn * Cn, 32, 0, stream>>>(X, meanOut, varPart);
  k_channel<<<Cn, 32, 0, stream>>>(meanOut, varPart, var, Rinv);
  k_rb<<<1, 32, 0, stream>>>(bias, Rb);
  // Stage 4: center -> gyro scale -> re-bias -> output.
  k_final<<<Bn * Cn, 32, 0, stream>>>(X, Rinv, Rb, var, shift, out);
}